// _GS_model_12979391168797
// MI455X (gfx1250) — compile-verified
//
#include <hip/hip_runtime.h>

// ---------------------------------------------------------------------------
// Gaussian splat forward render for MI455X (gfx1250, wave32).
//   K=16 footprint, IMG=1024 -> image (1025,1025,3) f32.
//   Screen tiled 16x16 -> 65x65 = 4225 tiles. Each gaussian overlaps <=4 tiles.
//   Deterministic chunked counting-sort gives per-tile lists ordered by
//   gaussian index (== reference's stable-sort composite order).
//   Compositing: 1 wave per tile; the 16x16 exponent field is a rank-3
//   outer product computed with V_WMMA_F32_16X16X4_F32 (log2-scaled so the
//   per-pixel gaussian value is a single v_exp_f32). Inner loop is fully
//   branchless: masked alpha=0 leaves T and color untouched.
// ---------------------------------------------------------------------------

#define TILE_W 65
#define NT (TILE_W * TILE_W)
#define NCHUNK 256
#define IMGW 1025 // W+1 == H+1
#define LOG2E 1.4426950408889634f

typedef __attribute__((ext_vector_type(2))) float v2f;
typedef __attribute__((ext_vector_type(8))) float v8f;

__global__ void zero_counts(int* __restrict__ counts, int m) {
  int i = blockIdx.x * blockDim.x + threadIdx.x;
  int stride = gridDim.x * blockDim.x;
  for (; i < m; i += stride) counts[i] = 0;
}

__global__ void count_kernel(const int* __restrict__ sp, int* __restrict__ counts,
                             int n, int chunkSize) {
  int i = blockIdx.x * blockDim.x + threadIdx.x;
  if (i >= n) return;
  int sx = sp[2 * i], sy = sp[2 * i + 1];
  int tx0 = sx >> 4, tx1 = (sx + 15) >> 4;
  int ty0 = sy >> 4, ty1 = (sy + 15) >> 4;
  int c = i / chunkSize;
  for (int ty = ty0; ty <= ty1; ++ty)
    for (int tx = tx0; tx <= tx1; ++tx)
      atomicAdd(&counts[(ty * TILE_W + tx) * NCHUNK + c], 1);
}

__global__ void tile_reduce(const int* __restrict__ counts, int* __restrict__ tile_total) {
  int t = blockIdx.x, c = threadIdx.x;
  __shared__ int s[NCHUNK];
  s[c] = counts[t * NCHUNK + c];
  __syncthreads();
  for (int off = NCHUNK / 2; off > 0; off >>= 1) {
    if (c < off) s[c] += s[c + off];
    __syncthreads();
  }
  if (c == 0) tile_total[t] = s[0];
}

__global__ void tile_scan(const int* __restrict__ tile_total, int* __restrict__ tile_start,
                          int nt) {
  __shared__ int s[1024];
  __shared__ int srun;
  int tid = threadIdx.x;
  if (tid == 0) srun = 0;
  __syncthreads();
  for (int base = 0; base < nt; base += 1024) {
    int i = base + tid;
    int v = (i < nt) ? tile_total[i] : 0;
    s[tid] = v;
    __syncthreads();
    int acc = v;
    for (int off = 1; off < 1024; off <<= 1) {
      int u = (tid >= off) ? s[tid - off] : 0;
      __syncthreads();
      acc += u;
      s[tid] = acc;
      __syncthreads();
    }
    int run = srun;
    if (i < nt) tile_start[i] = run + acc - v; // exclusive prefix
    __syncthreads();
    if (tid == 1023) srun = run + s[1023];
    __syncthreads();
  }
  if (tid == 0) tile_start[nt] = srun;
}

__global__ void chunk_scan(const int* __restrict__ counts, const int* __restrict__ tile_start,
                           int* __restrict__ cursors) {
  int t = blockIdx.x, c = threadIdx.x;
  __shared__ int s[NCHUNK];
  int v = counts[t * NCHUNK + c];
  s[c] = v;
  __syncthreads();
  int acc = v;
  for (int off = 1; off < NCHUNK; off <<= 1) {
    int u = (c >= off) ? s[c - off] : 0;
    __syncthreads();
    acc += u;
    s[c] = acc;
    __syncthreads();
  }
  cursors[t * NCHUNK + c] = tile_start[t] + acc - v;
}

// One thread per chunk, serial over its gaussians in index order -> each
// (tile,chunk) segment is filled in ascending gaussian index. Cells are
// private to a chunk, so plain load/store RMW is race-free & deterministic.
__global__ void fill_kernel(const int* __restrict__ sp, int* __restrict__ cursors,
                            int* __restrict__ entries, int n, int chunkSize) {
  int c = threadIdx.x; // single block of NCHUNK threads
  int i0 = c * chunkSize;
  int i1 = i0 + chunkSize;
  if (i1 > n) i1 = n;
  for (int i = i0; i < i1; ++i) {
    int sx = sp[2 * i], sy = sp[2 * i + 1];
    int tx0 = sx >> 4, tx1 = (sx + 15) >> 4;
    int ty0 = sy >> 4, ty1 = (sy + 15) >> 4;
    for (int ty = ty0; ty <= ty1; ++ty)
      for (int tx = tx0; tx <= tx1; ++tx) {
        int cell = (ty * TILE_W + tx) * NCHUNK + c;
        int pos = cursors[cell];
        cursors[cell] = pos + 1;
        entries[pos] = i;
      }
  }
}

// One wave32 per screen tile.
// WMMA A (16x4, M=x): lane l row M=l&15; VGPR0 = K0|K2, VGPR1 = K1|K3.
//   f = log2e * [-0.5a*dx^2, -0.5(b01+b10)*dx, -0.5c, 0]
// WMMA B (4x16, N=y): VGPR0 = row K0|K2, VGPR1 = row K1|K3, col N=l&15.
//   h = [1, dy, dy^2, 0]
// D[M=x][N=y] = f·h = log2-exponent. D layout: lane owns column-pair
//   x = r + 8*(lane>=16), y = lane&15  -> 8 pixels of state per lane.
__global__ void composite_kernel(const int* __restrict__ sp, const float* __restrict__ mean,
                                 const float* __restrict__ vinv, const float* __restrict__ opac,
                                 const float* __restrict__ ld,
                                 const int* __restrict__ tile_start,
                                 const int* __restrict__ entries, float* __restrict__ out) {
  int t = blockIdx.x;
  int lane = threadIdx.x;
  int tx = t % TILE_W, ty = t / TILE_W;
  int x0 = tx << 4, y0 = ty << 4;
  int half = lane >> 4;
  int l15 = lane & 15;
  int ypix = y0 + l15;

  float T[8], cr[8], cg[8], cb[8];
#pragma unroll
  for (int r = 0; r < 8; ++r) {
    T[r] = 1.0f;
    cr[r] = 0.0f;
    cg[r] = 0.0f;
    cb[r] = 0.0f;
  }

  int e0 = tile_start[t];
  int e1 = tile_start[t + 1];
  for (int e = e0; e < e1; ++e) {
    __builtin_prefetch(&entries[e + 16], 0, 0); // global_prefetch_b8
    int idx = entries[e];
    int sx = sp[2 * idx], sy = sp[2 * idx + 1];
    float mx = mean[2 * idx], my = mean[2 * idx + 1];
    float ka = -0.5f * LOG2E * vinv[4 * idx];
    float kb = -0.5f * LOG2E * (vinv[4 * idx + 1] + vinv[4 * idx + 2]);
    float kc = -0.5f * LOG2E * vinv[4 * idx + 3];
    float o = opac[idx];
    float lr = ld[3 * idx], lg = ld[3 * idx + 1], lb = ld[3 * idx + 2];

    float dx = (float)(x0 + l15) - mx;
    float dy = (float)ypix - my;

    v2f A, B;
    A.x = half ? kc : (ka * dx * dx);
    A.y = half ? 0.0f : (kb * dx);
    B.x = half ? (dy * dy) : 1.0f;
    B.y = half ? 0.0f : dy;
    v8f C = {0.f, 0.f, 0.f, 0.f, 0.f, 0.f, 0.f, 0.f};
    // D = A(16x4) x B(4x16) : full 16x16 log2-exponent field in one matrix op.
    v8f E = __builtin_amdgcn_wmma_f32_16x16x4_f32(false, A, false, B, (short)0, C,
                                                  false, false);

    bool inY = (ypix >= sy) && (ypix < sy + 16);
#pragma unroll
    for (int r = 0; r < 8; ++r) {
      int xr = x0 + r + (half << 3);
      bool in = inY && (xr >= sx) && (xr < sx + 16);
      float g = __builtin_amdgcn_exp2f(E[r]); // v_exp_f32
      float al = in ? (o * g) : 0.0f;         // branchless mask
      float w = T[r] * al;
      cr[r] += w * lr;
      cg[r] += w * lg;
      cb[r] += w * lb;
      T[r] *= (1.0f - al);
    }
  }

#pragma unroll
  for (int r = 0; r < 8; ++r) {
    int xr = x0 + r + (half << 3);
    if (xr < IMGW && ypix < IMGW) {
      size_t p = ((size_t)ypix * IMGW + xr) * 3;
      out[p] = cr[r];
      out[p + 1] = cg[r];
      out[p + 2] = cb[r];
    }
  }
}

extern "C" void kernel_launch(void* const* d_in, const int* in_sizes, int n_in, void* d_out,
                              int out_size, void* d_ws, size_t ws_size, hipStream_t stream) {
  const int* sp = (const int*)d_in[0];        // startpoint (N,2) i32
  const float* mean = (const float*)d_in[1];  // (N,2)
  const float* vinv = (const float*)d_in[2];  // (N,2,2)
  const float* opac = (const float*)d_in[3];  // (N,1)
  const float* ld = (const float*)d_in[4];    // (N,3)
  float* out = (float*)d_out;                 // (1025,1025,3)
  int n = in_sizes[0] / 2;
  int chunkSize = (n + NCHUNK - 1) / NCHUNK;

  char* ws = (char*)d_ws;
  size_t off = 0;
  int* counts = (int*)(ws + off);
  off += (size_t)NT * NCHUNK * sizeof(int);
  int* cursors = (int*)(ws + off);
  off += (size_t)NT * NCHUNK * sizeof(int);
  int* tile_total = (int*)(ws + off);
  off += (size_t)NT * sizeof(int);
  int* tile_start = (int*)(ws + off);
  off += (size_t)(NT + 1) * sizeof(int);
  int* entries = (int*)(ws + off);
  off += (size_t)4 * n * sizeof(int);

  zero_counts<<<512, 256, 0, stream>>>(counts, NT * NCHUNK);
  count_kernel<<<(n + 255) / 256, 256, 0, stream>>>(sp, counts, n, chunkSize);
  tile_reduce<<<NT, NCHUNK, 0, stream>>>(counts, tile_total);
  tile_scan<<<1, 1024, 0, stream>>>(tile_total, tile_start, NT);
  chunk_scan<<<NT, NCHUNK, 0, stream>>>(counts, tile_start, cursors);
  fill_kernel<<<1, NCHUNK, 0, stream>>>(sp, cursors, entries, n, chunkSize);
  composite_kernel<<<NT, 32, 0, stream>>>(sp, mean, vinv, opac, ld, tile_start, entries, out);
}